// Model_28518582845837
// MI455X (gfx1250) — compile-verified
//
#include <hip/hip_runtime.h>
#include <math.h>

typedef __attribute__((ext_vector_type(2))) float v2f;
typedef __attribute__((ext_vector_type(8))) float v8f;

#define SEQLEN  2304
#define WIN     48
#define RES     50
#define NSEG    48
#define WASH    10
#define KH      1900      // (48-10)*50, K of the big GEMM
#define NOUT    350       // 7*50
#define PREDLEN 336

// ---- LDS layout (in floats) ----
#define SZ_WPROJ (48*52)          // W_proj staged [n=48][k=52], k>=50 zero-padded
#define OFF_WPROJ 0
#define OFF_H     (OFF_WPROJ + SZ_WPROJ)      // h tile [16][1900]
#define SZ_H      (16*KH)
#define OFF_WIN   (OFF_H + SZ_H)              // W_in padded [64][48], rows >=50 zero (phase A only)
#define SZ_WINP   (64*WIN)                    // 3072 floats
#define OFF_Y1    OFF_WIN                     // y1 tile [16][354] (reuses W_in region; 5664 > 3072)
#define Y1S       354
#define SZ_Y1     (16*Y1S)
#define SMEM_FLOATS (OFF_Y1 + SZ_Y1)          // 38560 floats = 154,240 B (2 WGs/WGP)

__device__ __forceinline__ v8f wmma4(v2f a, v2f b, v8f c) {
    // V_WMMA_F32_16X16X4_F32 : fp32 A/B, fp32 accumulate
    return __builtin_amdgcn_wmma_f32_16x16x4_f32(false, a, false, b, (short)0, c, false, false);
}

__global__ __launch_bounds__(256)
void esn_fused(const float* __restrict__ x,
               const float* __restrict__ W_in,
               const float* __restrict__ W_out1,
               const float* __restrict__ W_proj,
               float* __restrict__ out)
{
    __shared__ float smem[SMEM_FLOATS];
    const int tid  = threadIdx.x;
    const int lane = tid & 31;
    const int wv   = tid >> 5;          // 8 waves of 32 (wave32)
    const int mrow = lane & 15;         // M row / N col within a 16-wide tile
    const int hf   = lane >> 4;         // half-wave selects K pair (ISA A/B layout)
    const int b0   = blockIdx.x * 16;   // 16 batch rows per workgroup

    // ---- stage small weights into LDS ----
    // W_in as [64][48], rows 50..63 zeroed -> branchless/mask-free B fragments in phase A
    for (int i = tid; i < SZ_WINP; i += 256)
        smem[OFF_WIN + i] = (i < RES * WIN) ? W_in[i] : 0.0f;
    // W_proj as [48][52], k=50,51 zeroed -> mask-free K padding in phase C
    for (int i = tid; i < SZ_WPROJ; i += 256) {
        int n = i / 52, k = i - n * 52;
        smem[OFF_WPROJ + i] = (k < RES) ? W_proj[n * RES + k] : 0.0f;
    }
    __syncthreads();

    // ================= Phase A: h = 0.5*tanh(xs @ W_in^T), segs 10..47 =================
    {
        const float* xrow = x + (size_t)(b0 + mrow) * SEQLEN + 2 * hf;
        for (int s = WASH + wv; s < NSEG; s += 8) {
            v8f c[4] = {};
            const float* xa = xrow + s * WIN;
            #pragma unroll
            for (int kk = 0; kk < 12; ++kk) {                // K = 48 = 12 x 4
                const int k0 = 4 * kk;
                const int kb = k0 + 2 * hf;
                v2f a = *(const v2f*)(xa + k0);
                #pragma unroll
                for (int nt = 0; nt < 4; ++nt) {
                    v2f b = *(const v2f*)&smem[OFF_WIN + (nt * 16 + mrow) * WIN + kb];
                    c[nt] = wmma4(a, b, c[nt]);
                }
            }
            const int so = (s - WASH) * RES;
            #pragma unroll
            for (int nt = 0; nt < 4; ++nt) {
                const int n = nt * 16 + mrow;
                if (n < RES) {
                    #pragma unroll
                    for (int j = 0; j < 8; ++j) {
                        const int m = j + 8 * hf;            // C/D layout: vgpr j -> M=j (+8 upper half)
                        smem[OFF_H + m * KH + so + n] = 0.5f * tanhf(c[nt][j]);
                    }
                }
            }
        }
    }
    __syncthreads();

    // ================= Phase B: y1 = h @ W_out1^T   [16 x 1900] x [1900 x 350] =================
    {
        const float* hrow = &smem[OFF_H + mrow * KH + 2 * hf];
        for (int nt2 = wv; nt2 < 22; nt2 += 8) {             // 350 -> 22 N-tiles
            const int nbase = nt2 * 16;
            const int n     = nbase + mrow;
            const int nld   = (n < NOUT) ? n : (NOUT - 1);   // clamp for safe global load
            const float* wrow = W_out1 + (size_t)nld * KH + 2 * hf;
            v8f c = {};
            #pragma unroll 5
            for (int kk = 0; kk < 475; ++kk) {               // K = 1900 = 475 x 4
                v2f a = *(const v2f*)(hrow + 4 * kk);        // ds_load_b64, 16-way bank-clean
                v2f b = *(const v2f*)(wrow + 4 * kk);        // streams from L2 (2.7MB shared by all WGs)
                c = wmma4(a, b, c);
            }
            const float msk = (n < NOUT) ? 1.0f : 0.0f;      // zero pad cols 350/351 (read by phase C)
            #pragma unroll
            for (int j = 0; j < 8; ++j) {
                const int m = j + 8 * hf;
                smem[OFF_Y1 + m * Y1S + nbase + mrow] = c[j] * msk;
            }
        }
    }
    __syncthreads();

    // ================= Phase C: out = y1.reshape(16,7,50) @ W_proj^T =================
    {
        for (int t = wv; t < 21; t += 8) {                   // 7 segs x 3 N-tiles
            const int seg = t / 3;
            const int nb  = (t - seg * 3) * 16;
            const float* arow = &smem[OFF_Y1 + mrow * Y1S + seg * RES + 2 * hf];
            const float* brow = &smem[OFF_WPROJ + (nb + mrow) * 52 + 2 * hf];
            v8f c = {};
            #pragma unroll
            for (int kk = 0; kk < 13; ++kk) {                // K = 50 -> 52 (B zero-padded)
                v2f a = *(const v2f*)(arow + 4 * kk);
                v2f b = *(const v2f*)(brow + 4 * kk);
                c = wmma4(a, b, c);
            }
            #pragma unroll
            for (int j = 0; j < 8; ++j) {
                const int m = j + 8 * hf;
                out[(size_t)(b0 + m) * PREDLEN + seg * WIN + nb + mrow] = c[j];
            }
        }
    }
}

extern "C" void kernel_launch(void* const* d_in, const int* in_sizes, int n_in,
                              void* d_out, int out_size, void* d_ws, size_t ws_size,
                              hipStream_t stream) {
    const float* x      = (const float*)d_in[0];
    const float* W_in   = (const float*)d_in[1];
    const float* W_out1 = (const float*)d_in[2];
    const float* W_proj = (const float*)d_in[3];
    float* out = (float*)d_out;
    const int B = in_sizes[0] / SEQLEN;      // 8192
    dim3 grid(B / 16);                       // one 16-batch M-tile per workgroup
    esn_fused<<<grid, 256, 0, stream>>>(x, W_in, W_out1, W_proj, out);
}